// PSROI_layer_50019189129760
// MI455X (gfx1250) — compile-verified
//
#include <hip/hip_runtime.h>
#include <hip/hip_bf16.h>

// PSROI pooling, MI455X (gfx1250, wave32).
// Stage 1: per-plane 2D integral image.
//   - Plane (80x80 f32, contiguous 25.6KB) DMA'd into LDS by the Tensor Data
//     Mover (TENSOR_LOAD_TO_LDS, iterate mode -> 81-dword padded rows, bank-
//     conflict-free strided A-fragment reads).
//   - cumsum_W = X*U and cumsum_H = L*Y as triangular-matrix GEMMs on
//     V_WMMA_F32_16X16X4_F32 (exact f32, matches reference arithmetic).
// Stage 2: 4-corner integral-image lookups per (box, channel); S (48MB) fits
//   in the 192MB L2 so the gathers are L2-resident.

#define ALPHA 5
#define PP    7
#define NB    8
#define CC    245           // ALPHA*PP*PP
#define HW    80
#define RR    8192
#define LDST  81            // padded LDS row stride (conflict-free A reads)

typedef __attribute__((ext_vector_type(2))) float v2f;
typedef __attribute__((ext_vector_type(8))) float v8f;
typedef __attribute__((ext_vector_type(4))) unsigned int v4u;
typedef __attribute__((ext_vector_type(8))) int v8i;
typedef __attribute__((ext_vector_type(4))) int v4i;

// ---------------------------------------------------------------------------
// Kernel 1: one block per (n,c) plane. 256 threads = 8 waves; 25 16x16 output
// tiles per pass, round-robined over waves via a *scalar* (readfirstlane)
// wave id so every loop around a WMMA is uniform (EXEC stays all-1s).
// ---------------------------------------------------------------------------
__global__ void __launch_bounds__(256)
integral_wmma_kernel(const float* __restrict__ fsam, float* __restrict__ S) {
    __shared__ float ldsX[HW * LDST];
    __shared__ float ldsY[HW * LDST];

    const int plane = blockIdx.x;                // 0 .. N*C-1
    const float* __restrict__ src = fsam + (size_t)plane * (HW * HW);
    float* __restrict__ dst       = S    + (size_t)plane * (HW * HW);

    const int wave_u = __builtin_amdgcn_readfirstlane((int)(threadIdx.x >> 5));
    const int lane = threadIdx.x & 31;
    const int m    = lane & 15;                  // A-row / B-col / D-col index
    const int kh   = (lane >> 4) << 1;           // K half select: 0 or 2
    const int drow = (lane >> 4) << 3;           // D-row offset: 0 or 8

    // ---- TDM: DMA the plane into LDS with an 81-dword padded row stride ----
    // D# per CDNA5 ISA ch.8: iterate mode, 80 iterations of one 80-elem row,
    // global advances 80 elems/iter, LDS advances 81 elems/iter (padding).
    if (wave_u == 0) {
        const unsigned int lds_off = (unsigned int)(size_t)&ldsX[0]; // LDS byte offset
        const unsigned long long ga = (unsigned long long)(const void*)src;
        v4u g0;
        g0[0] = 1u;                                   // count=1, user D#
        g0[1] = lds_off;                              // lds_addr [63:32]
        g0[2] = (unsigned int)(ga & 0xFFFFFFFFu);     // global_addr [95:64]
        g0[3] = (unsigned int)((ga >> 32) & 0x01FFFFFFu) | (2u << 30); // [120:96] | type=2
        v8i g1;
        g1[0] = (2 << 16) | (1 << 19);                // data_size=4B, iterate_enable=1
        g1[1] = (int)(80u << 16);                     // tensor_dim0=80  (bits 79:48)
        g1[2] = (int)(80u << 16);                     // tensor_dim1=80  (bits 111:80)
        g1[3] = (int)(80u << 16);                     // tile_dim0=80    (bits 127:112)
        g1[4] = 1;                                    // tile_dim1=1     (bits 143:128)
        g1[5] = 80;                                   // tensor_dim0_stride=80 (bits 207:160)
        g1[6] = 0;
        g1[7] = 0;
        v4i g2;
        g2[0] = 0;                                    // tensor_dim2 (unused)
        g2[1] = 81;                                   // lds_addr_increment = 81 elems
        g2[2] = 80;                                   // global_addr_increment = 80 elems
        g2[3] = (int)(79u << 16);                     // iterate_count = 79 (-> 80 iters)
        v4i g3 = {};
#if defined(__clang_major__) && (__clang_major__ >= 23)
        v8i gx = {};
        __builtin_amdgcn_tensor_load_to_lds(g0, g1, g2, g3, gx, 0);
#else
        __builtin_amdgcn_tensor_load_to_lds(g0, g1, g2, g3, 0);
#endif
        __builtin_amdgcn_s_wait_tensorcnt(0);
    }
    __syncthreads();

    // ---- Pass A: Y = X * U  (U[k,n] = 1 if k<=n)  == cumsum along W ----
    for (int t = wave_u; t < 25; t += 8) {
        const int ti = t / 5, tj = t - ti * 5;   // scalar
        v8f acc = {};
        const int kcmax = tj * 4 + 3;            // zero U-blocks skipped (uniform bound)
        for (int kc = 0; kc <= kcmax; ++kc) {
            const int kb = kc * 4 + kh;          // global K for this lane's pair
            v2f a, b;
            a.x = ldsX[(ti * 16 + m) * LDST + kb];
            a.y = ldsX[(ti * 16 + m) * LDST + kb + 1];
            const int n = tj * 16 + m;           // global column
            b.x = (kb     <= n) ? 1.0f : 0.0f;   // triangular ones, generated in regs
            b.y = (kb + 1 <= n) ? 1.0f : 0.0f;
            acc = __builtin_amdgcn_wmma_f32_16x16x4_f32(
                false, a, false, b, (short)0, acc, false, false);
        }
        const int rbase = ti * 16 + drow;
        const int col   = tj * 16 + m;
#pragma unroll
        for (int v = 0; v < 8; ++v)
            ldsY[(rbase + v) * LDST + col] = acc[v];
    }
    __syncthreads();

    // ---- Pass B: S = L * Y  (L[mg,k] = 1 if k<=mg)  == cumsum along H ----
    for (int t = wave_u; t < 25; t += 8) {
        const int ti = t / 5, tj = t - ti * 5;
        v8f acc = {};
        const int kcmax = ti * 4 + 3;            // zero L-blocks skipped (uniform bound)
        for (int kc = 0; kc <= kcmax; ++kc) {
            const int kb = kc * 4 + kh;
            v2f a, b;
            const int mg = ti * 16 + m;          // global output row
            a.x = (kb     <= mg) ? 1.0f : 0.0f;
            a.y = (kb + 1 <= mg) ? 1.0f : 0.0f;
            b.x = ldsY[(kb)     * LDST + tj * 16 + m];  // row-across-lanes: conflict-free
            b.y = ldsY[(kb + 1) * LDST + tj * 16 + m];
            acc = __builtin_amdgcn_wmma_f32_16x16x4_f32(
                false, a, false, b, (short)0, acc, false, false);
        }
        const int rbase = ti * 16 + drow;
        const int col   = tj * 16 + m;
#pragma unroll
        for (int v = 0; v < 8; ++v)
            dst[(rbase + v) * HW + col] = acc[v];
    }
}

// ---------------------------------------------------------------------------
// Kernel 2: one block per box (256 threads, thread c = output channel).
// Integral image stored unpadded; index 0 of the padded S is an implicit 0.
// ---------------------------------------------------------------------------
__device__ __forceinline__ float ii_lookup(const float* __restrict__ pl, int h, int w) {
    return (h == 0 || w == 0) ? 0.0f : pl[(h - 1) * HW + (w - 1)];
}

__global__ void __launch_bounds__(256)
psroi_pool_kernel(const float* __restrict__ box, const float* __restrict__ S,
                  float* __restrict__ out) {
    const int r = blockIdx.x;
    const int c = threadIdx.x;
    if (c >= CC) return;

    const float* bx = box + r * 5;               // same addr across block -> broadcast
    const int   b  = (int)bx[0];
    const float x1 = rintf(bx[1]);               // jnp.round = half-to-even = v_rndne
    const float y1 = rintf(bx[2]);
    const float x2 = rintf(bx[3] + 1.0f);
    const float y2 = rintf(bx[4] + 1.0f);
    const float bin_w = fmaxf(x2 - x1, 0.1f) * (1.0f / PP);
    const float bin_h = fmaxf(y2 - y1, 0.1f) * (1.0f / PP);

    const int pw = c % PP;
    const int ph = (c / PP) % PP;                // chan == c

    float g;
    g = floorf((float)ph * bin_h + y1);          const int hs = (int)fminf(fmaxf(g, 0.f), (float)HW);
    g = ceilf((float)(ph + 1) * bin_h + y1);     const int he = (int)fminf(fmaxf(g, 0.f), (float)HW);
    g = floorf((float)pw * bin_w + x1);          const int ws = (int)fminf(fmaxf(g, 0.f), (float)HW);
    g = ceilf((float)(pw + 1) * bin_w + x1);     const int we = (int)fminf(fmaxf(g, 0.f), (float)HW);

    const float* __restrict__ pl = S + ((size_t)b * CC + c) * (HW * HW);
    const float bin_sum = ii_lookup(pl, he, we) - ii_lookup(pl, hs, we)
                        - ii_lookup(pl, he, ws) + ii_lookup(pl, hs, ws);
    const int area = (he - hs) * (we - ws);
    out[(size_t)r * CC + c] = (area > 0) ? bin_sum / fmaxf((float)area, 1.0f) : 0.0f;
}

// ---------------------------------------------------------------------------
extern "C" void kernel_launch(void* const* d_in, const int* in_sizes, int n_in,
                              void* d_out, int out_size, void* d_ws, size_t ws_size,
                              hipStream_t stream) {
    const float* fsam = (const float*)d_in[0];   // (8, 245, 80, 80) f32
    const float* box  = (const float*)d_in[1];   // (8192, 5) f32
    float* out = (float*)d_out;                  // (8192, 245) f32
    float* S   = (float*)d_ws;                   // integral images: N*C*80*80 f32 (~48 MiB)

    integral_wmma_kernel<<<NB * CC, 256, 0, stream>>>(fsam, S);
    psroi_pool_kernel<<<RR, 256, 0, stream>>>(box, S, out);
}